// LinearSelector_87454124081772
// MI455X (gfx1250) — compile-verified
//
#include <hip/hip_runtime.h>

// Problem constants (match reference)
#define B_    32
#define LQ    32
#define DLEN  4096
#define P_    16
#define D_    2048   // embedding dim (= K of GEMM)
#define H_    2048   // output dim (= N of GEMM)
#define MTOT  544    // 32 q rows + 512 passage rows

typedef __attribute__((ext_vector_type(2))) float v2f;
typedef __attribute__((ext_vector_type(4))) float v4f;
typedef __attribute__((ext_vector_type(8))) float v8f;

// ---------------------------------------------------------------------------
// Kernel 1: q_rep[b][d] = (1/len) * sum_{l<len} emb[queries[b][l]][d]
// One block per batch row; 256 threads x 8 floats = 2048 columns.
// ---------------------------------------------------------------------------
__global__ void qrep_kernel(const int* __restrict__ queries,
                            const int* __restrict__ qlen,
                            const float* __restrict__ emb,
                            float* __restrict__ rep) {
    const int b  = blockIdx.x;
    const int d0 = threadIdx.x * 8;
    const int L  = qlen[b];
    v4f acc0 = {0.f, 0.f, 0.f, 0.f};
    v4f acc1 = {0.f, 0.f, 0.f, 0.f};
    for (int l = 0; l < L; ++l) {
        const int tok = queries[b * LQ + l];
        const v4f* e  = (const v4f*)(emb + (size_t)tok * D_ + d0);
        acc0 += e[0];
        acc1 += e[1];
    }
    const float inv = 1.0f / (float)L;
    v4f* out = (v4f*)(rep + (size_t)b * D_ + d0);
    out[0] = acc0 * inv;
    out[1] = acc1 * inv;
}

// ---------------------------------------------------------------------------
// Kernel 2: passage means. Passage p of batch b owns tokens
// [start, start+len) with start = sum_{j<p} plen[b][j] (inclusive-cumsum
// segmentation of the reference). One block per (b,p): streams len token
// rows (8KB each, fully coalesced: 256 threads x 32B), register accumulate.
// This is the bandwidth-dominant phase (~1 GiB of embedding reads).
// ---------------------------------------------------------------------------
__global__ void prep_kernel(const int* __restrict__ docs,
                            const int* __restrict__ plen,
                            const float* __restrict__ emb,
                            float* __restrict__ rep) {
    const int bp = blockIdx.x;          // 0..511
    const int b  = bp >> 4;
    const int p  = bp & 15;
    int start = 0;
    for (int j = 0; j < p; ++j) start += plen[b * P_ + j];
    const int len = plen[b * P_ + p];

    const int d0 = threadIdx.x * 8;
    v4f acc0 = {0.f, 0.f, 0.f, 0.f};
    v4f acc1 = {0.f, 0.f, 0.f, 0.f};
    const int* drow = docs + (size_t)b * DLEN + start;
    for (int i = 0; i < len; ++i) {
        const int tok = drow[i];        // scalar broadcast across the block
        const v4f* e  = (const v4f*)(emb + (size_t)tok * D_ + d0);
        acc0 += e[0];
        acc1 += e[1];
    }
    const float inv = 1.0f / (float)(len < 1 ? 1 : len);
    v4f* out = (v4f*)(rep + (size_t)(B_ + bp) * D_ + d0);
    out[0] = acc0 * inv;
    out[1] = acc1 * inv;
}

// ---------------------------------------------------------------------------
// Kernel 3: enc[m][h] = sum_k rep[m][k] * W[h][k] + bias[h]
// f32 WMMA (V_WMMA_F32_16X16X4_F32), register-blocked 2 M-tiles x 4 N-tiles
// per wave: per K-step load 2 A-frags + 4 B-frags (6 x 8B/lane), issue
// 8 WMMAs -> ~10.7 FLOP/B from L2 (vs 4 unblocked), 0.42 GB total traffic.
// A-frag (16x4 f32): lane ln supplies rep[mRow+ln][k0 + 2*half + {0,1}]
// B-frag (4x16, B[k][n] = W[n][k]): lane ln supplies W[nRow+ln][same k's]
// C/D: VGPR r holds D[r + 8*half][ln] within each 16x16 tile.
// M: 17 groups of 32 rows; N: 32 groups of 64 -> 544 waves = 68 blocks x 8.
// ---------------------------------------------------------------------------
__global__ void gemm_wmma_kernel(const float* __restrict__ rep,
                                 const float* __restrict__ W,
                                 const float* __restrict__ bias,
                                 float* __restrict__ enc) {
    const int wave = blockIdx.x * (blockDim.x >> 5) + (threadIdx.x >> 5); // 0..543
    const int lane = threadIdx.x & 31;
    const int half = lane >> 4;
    const int ln   = lane & 15;
    const int mg   = wave >> 5;   // 0..16  (M group: 2 tiles = 32 rows)
    const int ng   = wave & 31;   // 0..31  (N group: 4 tiles = 64 cols)

    const float* a0p = rep + (size_t)(mg * 32 + ln) * D_ + 2 * half;
    const float* a1p = a0p + 16 * (size_t)D_;
    const float* b0p = W + (size_t)(ng * 64 + ln) * D_ + 2 * half;
    const float* b1p = b0p + 16 * (size_t)D_;
    const float* b2p = b0p + 32 * (size_t)D_;
    const float* b3p = b0p + 48 * (size_t)D_;

    v8f c00 = {0.f,0.f,0.f,0.f,0.f,0.f,0.f,0.f}, c01 = c00, c02 = c00, c03 = c00;
    v8f c10 = c00, c11 = c00, c12 = c00, c13 = c00;

    for (int k = 0; k < D_; k += 4) {
        v2f a0 = *(const v2f*)(a0p + k);
        v2f a1 = *(const v2f*)(a1p + k);
        v2f b0 = *(const v2f*)(b0p + k);
        v2f b1 = *(const v2f*)(b1p + k);
        v2f b2 = *(const v2f*)(b2p + k);
        v2f b3 = *(const v2f*)(b3p + k);
        // (neg_a, A, neg_b, B, c_mod, C, reuse_a, reuse_b)
        c00 = __builtin_amdgcn_wmma_f32_16x16x4_f32(false, a0, false, b0, (short)0, c00, false, false);
        c01 = __builtin_amdgcn_wmma_f32_16x16x4_f32(false, a0, false, b1, (short)0, c01, false, false);
        c02 = __builtin_amdgcn_wmma_f32_16x16x4_f32(false, a0, false, b2, (short)0, c02, false, false);
        c03 = __builtin_amdgcn_wmma_f32_16x16x4_f32(false, a0, false, b3, (short)0, c03, false, false);
        c10 = __builtin_amdgcn_wmma_f32_16x16x4_f32(false, a1, false, b0, (short)0, c10, false, false);
        c11 = __builtin_amdgcn_wmma_f32_16x16x4_f32(false, a1, false, b1, (short)0, c11, false, false);
        c12 = __builtin_amdgcn_wmma_f32_16x16x4_f32(false, a1, false, b2, (short)0, c12, false, false);
        c13 = __builtin_amdgcn_wmma_f32_16x16x4_f32(false, a1, false, b3, (short)0, c13, false, false);
    }

    const v8f acc[2][4] = { {c00, c01, c02, c03}, {c10, c11, c12, c13} };
#pragma unroll
    for (int i = 0; i < 2; ++i) {
#pragma unroll
        for (int j = 0; j < 4; ++j) {
            const int   h  = ng * 64 + j * 16 + ln;
            const float bv = bias[h];
#pragma unroll
            for (int r = 0; r < 8; ++r) {
                const int m = mg * 32 + i * 16 + r + 8 * half;
                enc[(size_t)m * H_ + h] = acc[i][j][r] + bv;
            }
        }
    }
}

// ---------------------------------------------------------------------------
// Kernel 4: scores[b][p] = dot(enc_q[b], enc_p[b][p]) * (plen>0)
// One block per (b,p); 256-thread tree reduction over H=2048.
// ---------------------------------------------------------------------------
__global__ void score_kernel(const float* __restrict__ enc,
                             const int* __restrict__ plen,
                             float* __restrict__ out) {
    const int bp = blockIdx.x;
    const int b  = bp >> 4;
    const float* q  = enc + (size_t)b * H_;
    const float* pe = enc + (size_t)(B_ + bp) * H_;
    const int t = threadIdx.x;
    float s = 0.f;
    for (int h = t; h < H_; h += 256) s += q[h] * pe[h];
    __shared__ float red[256];
    red[t] = s;
    __syncthreads();
    for (int off = 128; off > 0; off >>= 1) {
        if (t < off) red[t] += red[t + off];
        __syncthreads();
    }
    if (t == 0) out[bp] = (plen[bp] > 0) ? red[0] : 0.0f;
}

// ---------------------------------------------------------------------------
extern "C" void kernel_launch(void* const* d_in, const int* in_sizes, int n_in,
                              void* d_out, int out_size, void* d_ws, size_t ws_size,
                              hipStream_t stream) {
    (void)in_sizes; (void)n_in; (void)out_size; (void)ws_size;
    const int*   queries = (const int*)d_in[0];
    const int*   qlen    = (const int*)d_in[1];
    const int*   docs    = (const int*)d_in[2];
    const int*   plen    = (const int*)d_in[3];
    const float* emb     = (const float*)d_in[4];
    const float* W       = (const float*)d_in[5];
    const float* bias    = (const float*)d_in[6];
    float*       out     = (float*)d_out;

    // workspace: rep[544][2048] then enc[544][2048]  (8.9 MB total)
    float* rep = (float*)d_ws;
    float* enc = rep + (size_t)MTOT * D_;

    qrep_kernel<<<B_,       256, 0, stream>>>(queries, qlen, emb, rep);
    prep_kernel<<<B_ * P_,  256, 0, stream>>>(docs, plen, emb, rep);
    gemm_wmma_kernel<<<68,  256, 0, stream>>>(rep, W, bias, enc);
    score_kernel<<<B_ * P_, 256, 0, stream>>>(enc, plen, out);
}